// EvoformerLiteBlock_58574763983765
// MI455X (gfx1250) — compile-verified
//
#include <hip/hip_runtime.h>
#include <math.h>

// ---------------------------------------------------------------------------
// EvoformerLite block for MI455X (gfx1250, wave32).
// - All GEMMs / attention contractions on V_WMMA_F32_16X16X4_F32 (fp32 matrix
//   pipe, bit-faithful to the fp32 reference).
// - GEMM weight panels staged into LDS by the Tensor Data Mover
//   (tensor_load_to_lds + s_wait_tensorcnt), shared by 4 waves per block.
// WMMA fragment layouts per CDNA5 ISA 7.12.2:
//   A (16x4 f32):  lane l: M = l%16;  VGPR0 = K=(l/16)*2, VGPR1 = +1
//   B (4x16 f32):  lane l: N = l%16;  VGPR0 = K=(l/16)*2, VGPR1 = +1
//   C/D (16x16):   lane l, VGPR r: M = r + (l/16)*8, N = l%16
// ---------------------------------------------------------------------------

typedef __attribute__((ext_vector_type(2))) float v2f;
typedef __attribute__((ext_vector_type(8))) float v8f;
typedef __attribute__((ext_vector_type(4))) unsigned int u32x4;
typedef __attribute__((ext_vector_type(8))) int i32x8;
typedef __attribute__((ext_vector_type(4))) int i32x4;

#define DMODEL 128
#define NHEAD  8
#define DHEAD  16
#define NEG_INF (-3.4028234663852886e38f)

#if __has_builtin(__builtin_amdgcn_tensor_load_to_lds)
#define HAVE_TDM 1
#else
#define HAVE_TDM 0
#endif

__device__ __forceinline__ float wave_sum(float v) {
#pragma unroll
  for (int off = 16; off >= 1; off >>= 1) v += __shfl_xor(v, off, 32);
  return v;
}

// ---------------------------------------------------------------------------
// LayerNorm over last dim (128). 8 tokens per 256-thread block, 1 wave/token.
// tr_n > 0: input token m = n*tr_l + l  -> output token l*tr_n + n (transpose)
// ---------------------------------------------------------------------------
__global__ void ln_kernel(const float* __restrict__ x, const float* __restrict__ g,
                          const float* __restrict__ b, float* __restrict__ out,
                          int M, int tr_l, int tr_n) {
  int wave = threadIdx.x >> 5;
  int lane = threadIdx.x & 31;
  int m = blockIdx.x * 8 + wave;
  if (m >= M) return;
  const float* xp = x + (size_t)m * DMODEL;
  float v0 = xp[lane], v1 = xp[lane + 32], v2 = xp[lane + 64], v3 = xp[lane + 96];
  float mean = wave_sum(v0 + v1 + v2 + v3) * (1.0f / 128.0f);
  float d0 = v0 - mean, d1 = v1 - mean, d2 = v2 - mean, d3 = v3 - mean;
  float var = wave_sum(d0 * d0 + d1 * d1 + d2 * d2 + d3 * d3) * (1.0f / 128.0f);
  float inv = rsqrtf(var + 1e-5f);
  int mo = (tr_n > 0) ? (m % tr_l) * tr_n + (m / tr_l) : m;
  float* op = out + (size_t)mo * DMODEL;
  op[lane]      = d0 * inv * g[lane]      + b[lane];
  op[lane + 32] = d1 * inv * g[lane + 32] + b[lane + 32];
  op[lane + 64] = d2 * inv * g[lane + 64] + b[lane + 64];
  op[lane + 96] = d3 * inv * g[lane + 96] + b[lane + 96];
}

// ---------------------------------------------------------------------------
// C[m,n] = act(A[m,:] @ W[:,n] + bias[n]) (+ resid).
// Block = 4 waves covering a 64-row M-strip at one 16-col N tile.
// The K x 16 weight panel is DMAed into LDS once per block by the TDM
// (tensor_load_to_lds, TENSORcnt) and shared by all 4 waves; per-lane B
// fragments then come from LDS (ds_load), A fragments from global (b64).
// map_n > 0: store row index remapped mo = (m % map_n)*map_l + m/map_n.
// ---------------------------------------------------------------------------
__global__ void gemm16_kernel(const float* __restrict__ A, const float* __restrict__ W,
                              const float* __restrict__ bias, const float* __restrict__ resid,
                              float* __restrict__ out, int M, int N, int K,
                              int act_gelu, int map_l, int map_n) {
  __shared__ float wpan[512 * 16];   // K <= 512 -> up to 32 KB
  int lane = threadIdx.x & 31;
  int wave = threadIdx.x >> 5;       // 0..3
  int row  = lane & 15;
  int half = lane >> 4;
  int n0 = blockIdx.x * 16;
  int m0 = blockIdx.y * 64 + wave * 16;

#if HAVE_TDM
  if (threadIdx.x == 0) {
    unsigned long long ga = (unsigned long long)(const void*)(W + n0);
    unsigned int lds_off = (unsigned int)(unsigned long long)(const void*)&wpan[0];
    u32x4 g0;
    g0[0] = 1u;                                            // count=1 (valid D#)
    g0[1] = lds_off;                                       // lds_addr (bytes)
    g0[2] = (unsigned int)(ga & 0xFFFFFFFFull);            // global_addr[31:0]
    g0[3] = (unsigned int)((ga >> 32) & 0x1FFFFFFull)      // global_addr[56:32]
            | (2u << 30);                                  // type=2 (image)
    i32x8 g1;
    g1[0] = (int)(2u << 16);                               // data_size=4B
    g1[1] = (int)(((unsigned int)N & 0xFFFFu) << 16);      // tensor_dim0[15:0] @48
    g1[2] = (int)((((unsigned int)N >> 16) & 0xFFFFu)      // tensor_dim0[31:16]
            | (((unsigned int)K & 0xFFFFu) << 16));        // tensor_dim1[15:0]
    g1[3] = (int)((((unsigned int)K >> 16) & 0xFFFFu)      // tensor_dim1[31:16]
            | (16u << 16));                                // tile_dim0 = 16
    g1[4] = (int)((unsigned int)K & 0xFFFFu);              // tile_dim1 = K
    g1[5] = (int)(unsigned int)N;                          // dim0_stride[31:0] = N
    g1[6] = 0;                                             // stride hi / dim1_stride
    g1[7] = 0;
    i32x4 gz = {0, 0, 0, 0};
    i32x8 gz8 = {0, 0, 0, 0, 0, 0, 0, 0};
    __builtin_amdgcn_tensor_load_to_lds(g0, g1, gz, gz, gz8, 0);
  }
  __builtin_amdgcn_s_wait_tensorcnt(0);
#else
  for (int idx = threadIdx.x; idx < K * 16; idx += 128) {
    wpan[idx] = W[(size_t)(idx >> 4) * N + n0 + (idx & 15)];
  }
#endif
  __syncthreads();

  const float* arow = A + (size_t)(m0 + row) * K;
  v8f acc = {};
  for (int k0 = 0; k0 < K; k0 += 4) {
    int ka = k0 + half * 2;
    v2f av; av.x = arow[ka];              av.y = arow[ka + 1];
    v2f bv; bv.x = wpan[ka * 16 + row];   bv.y = wpan[(ka + 1) * 16 + row];
    acc = __builtin_amdgcn_wmma_f32_16x16x4_f32(false, av, false, bv,
                                                (short)0, acc, false, false);
  }
  int n = n0 + row;
  float bsc = bias ? bias[n] : 0.0f;
#pragma unroll
  for (int r = 0; r < 8; ++r) {
    int m = m0 + r + half * 8;
    int mo = (map_n > 0) ? (m % map_n) * map_l + (m / map_n) : m;
    float v = acc[r] + bsc;
    if (act_gelu) v = 0.5f * v * (1.0f + erff(v * 0.70710678118654752f));
    if (resid) v += resid[(size_t)mo * N + n];
    out[(size_t)mo * N + n] = v;
  }
}

// ---------------------------------------------------------------------------
// Self-attention, one wave per (sequence, head, 16-query tile).
// qkv: [nseq*S, 384] packed (Q|K|V). Scores staged in LDS (16 x S), softmax
// done by the wave (2 lanes per score row), then P @ V via WMMA.
// ---------------------------------------------------------------------------
__global__ void attn_kernel(const float* __restrict__ qkv, float* __restrict__ o,
                            const unsigned char* __restrict__ mask, int S) {
  __shared__ float sc[16 * 512];
  int lane = threadIdx.x & 31;
  int row  = lane & 15;
  int half = lane >> 4;
  int seq = blockIdx.x;
  int h   = blockIdx.y;
  int qt  = blockIdx.z;
  size_t base = (size_t)seq * S;
  int qoff = h * DHEAD;

  v2f qa[4];
  {
    const float* qrow = qkv + (base + qt * 16 + row) * 384 + qoff;
#pragma unroll
    for (int kk = 0; kk < 4; ++kk) {
      int ka = kk * 4 + half * 2;
      qa[kk].x = qrow[ka];
      qa[kk].y = qrow[ka + 1];
    }
  }

  int nt = S >> 4;
  // --- scores = (Q @ K^T) * 1/sqrt(dh), masked, into LDS ---
  for (int t = 0; t < nt; ++t) {
    v8f s = {};
    const float* krow = qkv + (base + t * 16 + row) * 384 + 128 + qoff;
#pragma unroll
    for (int kk = 0; kk < 4; ++kk) {
      int ka = kk * 4 + half * 2;
      v2f bv; bv.x = krow[ka]; bv.y = krow[ka + 1];
      s = __builtin_amdgcn_wmma_f32_16x16x4_f32(false, qa[kk], false, bv,
                                                (short)0, s, false, false);
    }
    int key = t * 16 + row;
    bool masked = mask ? (mask[key] != 0) : false;
#pragma unroll
    for (int r = 0; r < 8; ++r) {
      float v = masked ? NEG_INF : s[r] * 0.25f;   // 1/sqrt(16)
      sc[(r + half * 8) * S + key] = v;
    }
  }

  // --- softmax over S per score row (lanes l and l+16 share row l%16) ---
  {
    int hs = S >> 1;
    float* rp = sc + row * S + half * hs;
    float mx = NEG_INF;
    for (int j = 0; j < hs; ++j) mx = fmaxf(mx, rp[j]);
    mx = fmaxf(mx, __shfl_xor(mx, 16, 32));
    float sum = 0.0f;
    for (int j = 0; j < hs; ++j) { float e = expf(rp[j] - mx); rp[j] = e; sum += e; }
    sum += __shfl_xor(sum, 16, 32);
    float inv = 1.0f / sum;
    for (int j = 0; j < hs; ++j) rp[j] *= inv;
  }

  // --- O = P @ V ---
  int voff = 256 + qoff;
  v8f acc = {};
  for (int t = 0; t < nt; ++t) {
#pragma unroll
    for (int kk = 0; kk < 4; ++kk) {
      int ka = kk * 4 + half * 2;
      v2f av; av.x = sc[row * S + t * 16 + ka];
              av.y = sc[row * S + t * 16 + ka + 1];
      const float* vp = qkv + (base + t * 16 + ka) * 384 + voff + row;
      v2f bv; bv.x = vp[0]; bv.y = vp[384];
      acc = __builtin_amdgcn_wmma_f32_16x16x4_f32(false, av, false, bv,
                                                  (short)0, acc, false, false);
    }
  }
#pragma unroll
  for (int r = 0; r < 8; ++r) {
    o[(base + qt * 16 + r + half * 8) * DMODEL + qoff + row] = acc[r];
  }
}

// residue[i] += mean over N rows of msa[n, i]
__global__ void pool_add_kernel(const float* __restrict__ msa, const float* __restrict__ resin,
                                float* __restrict__ out, int nrows, int ld) {
  int i = blockIdx.x * blockDim.x + threadIdx.x;
  if (i >= ld) return;
  float s = 0.0f;
  for (int n = 0; n < nrows; ++n) s += msa[(size_t)n * ld + i];
  out[i] = resin[i] + s * (1.0f / (float)nrows);
}

// ---------------------------------------------------------------------------
extern "C" void kernel_launch(void* const* d_in, const int* in_sizes, int n_in,
                              void* d_out, int out_size, void* d_ws, size_t ws_size,
                              hipStream_t stream) {
  (void)in_sizes; (void)n_in; (void)out_size; (void)ws_size;
  const int L = 512, Nrows = 64;
  const int Mmsa = Nrows * L;                 // 32768 tokens

  const float* msa_in = (const float*)d_in[0];
  const float* res_in = (const float*)d_in[1];
  const unsigned char* pad = (const unsigned char*)d_in[2];
  // params in JAX tree_leaves order: dicts flattened with sorted keys.
  // top: col(3..8), res(9..14), res_ffn(15..20), row(21..26), trans(27..32)
  // attn sub-order:  ln_b, ln_g, out_b, out_w, qkv_b, qkv_w
  // ffn  sub-order:  b1, b2, ln_b, ln_g, w1, w2
  const float* col_ln_b  = (const float*)d_in[3];
  const float* col_ln_g  = (const float*)d_in[4];
  const float* col_out_b = (const float*)d_in[5];
  const float* col_out_w = (const float*)d_in[6];
  const float* col_qkv_b = (const float*)d_in[7];
  const float* col_qkv_w = (const float*)d_in[8];
  const float* res_ln_b  = (const float*)d_in[9];
  const float* res_ln_g  = (const float*)d_in[10];
  const float* res_out_b = (const float*)d_in[11];
  const float* res_out_w = (const float*)d_in[12];
  const float* res_qkv_b = (const float*)d_in[13];
  const float* res_qkv_w = (const float*)d_in[14];
  const float* rf_b1     = (const float*)d_in[15];
  const float* rf_b2     = (const float*)d_in[16];
  const float* rf_ln_b   = (const float*)d_in[17];
  const float* rf_ln_g   = (const float*)d_in[18];
  const float* rf_w1     = (const float*)d_in[19];
  const float* rf_w2     = (const float*)d_in[20];
  const float* row_ln_b  = (const float*)d_in[21];
  const float* row_ln_g  = (const float*)d_in[22];
  const float* row_out_b = (const float*)d_in[23];
  const float* row_out_w = (const float*)d_in[24];
  const float* row_qkv_b = (const float*)d_in[25];
  const float* row_qkv_w = (const float*)d_in[26];
  const float* tr_b1     = (const float*)d_in[27];
  const float* tr_b2     = (const float*)d_in[28];
  const float* tr_ln_b   = (const float*)d_in[29];
  const float* tr_ln_g   = (const float*)d_in[30];
  const float* tr_w1     = (const float*)d_in[31];
  const float* tr_w2     = (const float*)d_in[32];

  float* MSA = (float*)d_out;                       // [64*512, 128]
  float* RES = MSA + (size_t)Mmsa * DMODEL;         // [512, 128]

  float* ws  = (float*)d_ws;
  float* lnb  = ws;                                 // 32768*128 floats
  float* big  = ws + (size_t)Mmsa * DMODEL;         // 32768*512 floats (reused)
  float* qkvb = big;                                // 32768*384
  float* ob   = big + (size_t)Mmsa * 384;           // 32768*128
  float* hb   = big;                                // 32768*512 (aliases qkv+o, later)

  dim3 b32(32), b128(128), b256(256);

  // ---- row self-attention (64 seqs of length 512, key padding mask) ----
  ln_kernel<<<Mmsa / 8, b256, 0, stream>>>(msa_in, row_ln_g, row_ln_b, lnb, Mmsa, 0, 0);
  gemm16_kernel<<<dim3(24, Mmsa / 64), b128, 0, stream>>>(lnb, row_qkv_w, row_qkv_b,
                                                          nullptr, qkvb, Mmsa, 384, 128, 0, 0, 0);
  attn_kernel<<<dim3(Nrows, NHEAD, L / 16), b32, 0, stream>>>(qkvb, ob, pad, L);
  gemm16_kernel<<<dim3(8, Mmsa / 64), b128, 0, stream>>>(ob, row_out_w, row_out_b,
                                                         msa_in, MSA, Mmsa, 128, 128, 0, 0, 0);

  // ---- col self-attention (512 seqs of length 64, LN fused w/ transpose) ----
  ln_kernel<<<Mmsa / 8, b256, 0, stream>>>(MSA, col_ln_g, col_ln_b, lnb, Mmsa, L, Nrows);
  gemm16_kernel<<<dim3(24, Mmsa / 64), b128, 0, stream>>>(lnb, col_qkv_w, col_qkv_b,
                                                          nullptr, qkvb, Mmsa, 384, 128, 0, 0, 0);
  attn_kernel<<<dim3(L, NHEAD, Nrows / 16), b32, 0, stream>>>(qkvb, ob, nullptr, Nrows);
  // output projection scatters back to (n,l)-major: mo = (m%64)*512 + m/64
  gemm16_kernel<<<dim3(8, Mmsa / 64), b128, 0, stream>>>(ob, col_out_w, col_out_b,
                                                         MSA, MSA, Mmsa, 128, 128, 0, L, Nrows);

  // ---- MSA transition FFN ----
  ln_kernel<<<Mmsa / 8, b256, 0, stream>>>(MSA, tr_ln_g, tr_ln_b, lnb, Mmsa, 0, 0);
  gemm16_kernel<<<dim3(32, Mmsa / 64), b128, 0, stream>>>(lnb, tr_w1, tr_b1,
                                                          nullptr, hb, Mmsa, 512, 128, 1, 0, 0);
  gemm16_kernel<<<dim3(8, Mmsa / 64), b128, 0, stream>>>(hb, tr_w2, tr_b2,
                                                         MSA, MSA, Mmsa, 128, 512, 0, 0, 0);

  // ---- residue track ----
  pool_add_kernel<<<(L * DMODEL + 255) / 256, b256, 0, stream>>>(MSA, res_in, RES, Nrows, L * DMODEL);
  ln_kernel<<<L / 8, b256, 0, stream>>>(RES, res_ln_g, res_ln_b, lnb, L, 0, 0);
  gemm16_kernel<<<dim3(24, L / 64), b128, 0, stream>>>(lnb, res_qkv_w, res_qkv_b,
                                                       nullptr, qkvb, L, 384, 128, 0, 0, 0);
  attn_kernel<<<dim3(1, NHEAD, L / 16), b32, 0, stream>>>(qkvb, ob, pad, L);
  gemm16_kernel<<<dim3(8, L / 64), b128, 0, stream>>>(ob, res_out_w, res_out_b,
                                                      RES, RES, L, 128, 128, 0, 0, 0);
  ln_kernel<<<L / 8, b256, 0, stream>>>(RES, rf_ln_g, rf_ln_b, lnb, L, 0, 0);
  gemm16_kernel<<<dim3(32, L / 64), b128, 0, stream>>>(lnb, rf_w1, rf_b1,
                                                       nullptr, hb, L, 512, 128, 1, 0, 0);
  gemm16_kernel<<<dim3(8, L / 64), b128, 0, stream>>>(hb, rf_w2, rf_b2,
                                                      RES, RES, L, 128, 512, 0, 0, 0);
}